// GCNWithEdgeFeatures_34230889349202
// MI455X (gfx1250) — compile-verified
//
#include <hip/hip_runtime.h>
#include <hip/hip_bf16.h>

typedef __attribute__((ext_vector_type(2))) float v2f;
typedef __attribute__((ext_vector_type(8))) float v8f;

#define THREADS 256

// ---------------- utility ----------------
__global__ void zero_kernel(float* __restrict__ p, long long n) {
  long long i = (long long)blockIdx.x * blockDim.x + threadIdx.x;
  if (i < n) p[i] = 0.0f;
}

// ---------------- edge fc: w[e] = edge_feats[e,:16] @ W_ef + b_ef ----------------
__global__ void edge_fc_kernel(const float* __restrict__ ef, const float* __restrict__ Wef,
                               const float* __restrict__ bef, float* __restrict__ w, int E) {
  int e = blockIdx.x * blockDim.x + threadIdx.x;
  if (e >= E) return;
  const float4* p = (const float4*)(ef + (size_t)e * 16);
  const float4* q = (const float4*)Wef;
  float acc = bef[0];
#pragma unroll
  for (int j = 0; j < 4; ++j) {
    float4 a = p[j];
    float4 b = q[j];
    acc += a.x * b.x + a.y * b.y + a.z * b.z + a.w * b.w;
  }
  w[e] = acc;
}

// ---------------- scatter weighted sums + unweighted degrees ----------------
__global__ void edge_degree_kernel(const float* __restrict__ w, const int* __restrict__ src,
                                   const int* __restrict__ dst,
                                   float* __restrict__ sum_out, float* __restrict__ sum_in,
                                   float* __restrict__ deg_out, float* __restrict__ deg_in, int E) {
  int e = blockIdx.x * blockDim.x + threadIdx.x;
  if (e >= E) return;
  float we = w[e];
  int s = src[e], d = dst[e];
  unsafeAtomicAdd(&sum_out[s], we);
  unsafeAtomicAdd(&sum_in[d], we);
  unsafeAtomicAdd(&deg_out[s], 1.0f);
  unsafeAtomicAdd(&deg_in[d], 1.0f);
}

// ---------------- deg -> 1/sqrt(max(deg,1)) in place ----------------
__global__ void node_norm_kernel(float* __restrict__ deg_out, float* __restrict__ deg_in, int N) {
  int i = blockIdx.x * blockDim.x + threadIdx.x;
  if (i >= N) return;
  deg_out[i] = 1.0f / sqrtf(fmaxf(deg_out[i], 1.0f));
  deg_in[i]  = 1.0f / sqrtf(fmaxf(deg_in[i], 1.0f));
}

// ---------------- w[e] /= sqrt(sum_out[src]*sum_in[dst]) ----------------
__global__ void edge_norm_kernel(float* __restrict__ w, const int* __restrict__ src,
                                 const int* __restrict__ dst,
                                 const float* __restrict__ sum_out,
                                 const float* __restrict__ sum_in, int E) {
  int e = blockIdx.x * blockDim.x + threadIdx.x;
  if (e >= E) return;
  w[e] = w[e] / sqrtf(sum_out[src[e]] * sum_in[dst[e]]);
}

// ---------------- GEMM: Y = (diag(rowscale) * X) @ W, K = Nout = 128 ----------------
// One wave computes one 16x16 output tile with V_WMMA_F32_16X16X4_F32.
// Rows must be a multiple of 16 (100000 = 6250*16). 8 waves/block cover the
// 8 column tiles of one row tile (shared A rows -> cache friendly).
__global__ void gemm_rowscale_wmma(const float* __restrict__ X, const float* __restrict__ W,
                                   const float* __restrict__ rowscale, float* __restrict__ Y,
                                   int nRowTiles) {
  int wave = threadIdx.x >> 5;
  int lane = threadIdx.x & 31;
  int tm = blockIdx.x;          // row tile
  int tn = wave;                // column tile 0..7
  if (tm >= nRowTiles) return;  // wave-uniform; EXEC stays all-1s for WMMA

  int r = lane & 15;            // row within tile (A) / col within tile (B,C,D)
  int klo = (lane >> 4) * 2;    // K sub-offset per half-wave
  int row = tm * 16 + r;
  int col = tn * 16 + r;

  float s = rowscale[row];
  const float* xr = X + (size_t)row * 128;

  v8f acc = {};
#pragma unroll
  for (int k = 0; k < 128; k += 4) {
    v2f a, b;
    a.x = xr[k + klo] * s;
    a.y = xr[k + klo + 1] * s;
    b.x = W[(size_t)(k + klo) * 128 + col];
    b.y = W[(size_t)(k + klo + 1) * 128 + col];
    acc = __builtin_amdgcn_wmma_f32_16x16x4_f32(
        /*neg_a=*/false, a, /*neg_b=*/false, b,
        /*c_mod=*/(short)0, acc, /*reuse_a=*/false, /*reuse_b=*/false);
  }

  // D layout: element i of lane holds (M = i + 8*(lane>>4), N = lane&15)
  int mbase = tm * 16 + 8 * (lane >> 4);
#pragma unroll
  for (int i = 0; i < 8; ++i) {
    Y[(size_t)(mbase + i) * 128 + col] = acc[i];
  }
}

// ---------------- edge aggregate: agg[dst] += h[src] * w[e] ----------------
// One wave per edge; each lane handles 4 contiguous floats (float4 load + 4 atomics).
__global__ void edge_aggregate_kernel(const float* __restrict__ H, const float* __restrict__ w,
                                      const int* __restrict__ src, const int* __restrict__ dst,
                                      float* __restrict__ out, int E) {
  long long t = (long long)blockIdx.x * blockDim.x + threadIdx.x;
  int e = (int)(t >> 5);
  if (e >= E) return;
  int lane = (int)(t & 31);
  float we = w[e];
  int s = src[e], d = dst[e];
  float4 v = ((const float4*)(H + (size_t)s * 128))[lane];
  float* op = out + (size_t)d * 128 + lane * 4;
  unsafeAtomicAdd(op + 0, v.x * we);
  unsafeAtomicAdd(op + 1, v.y * we);
  unsafeAtomicAdd(op + 2, v.z * we);
  unsafeAtomicAdd(op + 3, v.w * we);
}

// ---------------- finalize: x = [relu](x * norm_in[node] + b[feat]) ----------------
__global__ void finalize_kernel(float* __restrict__ x, const float* __restrict__ norm_in,
                                const float* __restrict__ b, long long total, int do_relu) {
  long long t = (long long)blockIdx.x * blockDim.x + threadIdx.x;
  if (t >= total) return;
  int node = (int)(t >> 7);
  int f = (int)(t & 127);
  float v = x[t] * norm_in[node] + b[f];
  x[t] = do_relu ? fmaxf(v, 0.0f) : v;
}

extern "C" void kernel_launch(void* const* d_in, const int* in_sizes, int n_in,
                              void* d_out, int out_size, void* d_ws, size_t ws_size,
                              hipStream_t stream) {
  const float* node_feats = (const float*)d_in[0];
  const float* edge_feats = (const float*)d_in[1];
  const float* W_ef = (const float*)d_in[2];
  const float* b_ef = (const float*)d_in[3];
  const float* W1 = (const float*)d_in[4];
  const float* b1 = (const float*)d_in[5];
  const float* W2 = (const float*)d_in[6];
  const float* b2 = (const float*)d_in[7];
  const int* src = (const int*)d_in[8];
  const int* dst = (const int*)d_in[9];

  const int N = in_sizes[0] / 128;
  const int E = in_sizes[8];
  const long long NF = (long long)N * 128;

  // workspace layout (floats)
  float* ws = (float*)d_ws;
  float* sum_out = ws;                       // [N]
  float* sum_in  = ws + (size_t)N;           // [N]
  float* nrm_out = ws + 2 * (size_t)N;       // [N] deg_out -> norm_out
  float* nrm_in  = ws + 3 * (size_t)N;       // [N] deg_in  -> norm_in
  float* w       = ws + 4 * (size_t)N;       // [E]
  float* h       = w + (size_t)E;            // [N,128]
  float* agg     = h + (size_t)N * 128;      // [N,128]
  float* out     = (float*)d_out;            // [N,128]

  const int eBlocks = (E + THREADS - 1) / THREADS;
  const int nBlocks = (N + THREADS - 1) / THREADS;
  const int nfBlocks = (int)((NF + THREADS - 1) / THREADS);
  const int aggBlocks = (int)(((long long)E * 32 + THREADS - 1) / THREADS);
  const int rowTiles = N / 16;               // 6250 (N is a multiple of 16)

  // --- edge weights + normalization terms ---
  zero_kernel<<<(4 * N + THREADS - 1) / THREADS, THREADS, 0, stream>>>(sum_out, 4LL * N);
  edge_fc_kernel<<<eBlocks, THREADS, 0, stream>>>(edge_feats, W_ef, b_ef, w, E);
  edge_degree_kernel<<<eBlocks, THREADS, 0, stream>>>(w, src, dst, sum_out, sum_in, nrm_out, nrm_in, E);
  node_norm_kernel<<<nBlocks, THREADS, 0, stream>>>(nrm_out, nrm_in, N);
  edge_norm_kernel<<<eBlocks, THREADS, 0, stream>>>(w, src, dst, sum_out, sum_in, E);

  // --- layer 1 ---
  gemm_rowscale_wmma<<<rowTiles, THREADS, 0, stream>>>(node_feats, W1, nrm_out, h, rowTiles);
  zero_kernel<<<nfBlocks, THREADS, 0, stream>>>(agg, NF);
  edge_aggregate_kernel<<<aggBlocks, THREADS, 0, stream>>>(h, w, src, dst, agg, E);
  finalize_kernel<<<nfBlocks, THREADS, 0, stream>>>(agg, nrm_in, b1, NF, /*relu=*/1);

  // --- layer 2 ---
  gemm_rowscale_wmma<<<rowTiles, THREADS, 0, stream>>>(agg, W2, nrm_out, h, rowTiles);
  zero_kernel<<<nfBlocks, THREADS, 0, stream>>>(out, NF);
  edge_aggregate_kernel<<<aggBlocks, THREADS, 0, stream>>>(h, w, src, dst, out, E);
  finalize_kernel<<<nfBlocks, THREADS, 0, stream>>>(out, nrm_in, b2, NF, /*relu=*/0);
}